// GCN_4406636445724
// MI455X (gfx1250) — compile-verified
//
#include <hip/hip_runtime.h>

// ---------------------------------------------------------------------------
// GCN 2-layer forward for MI455X (gfx1250, wave32).
//   g1  = Âx W1-projected:  h1 = x @ W1 ; g1 = Â @ h1
//   h1r = relu(g1 + b1)
//   g2  = Â @ h1r                      (associativity: (Â h1r) W2 == Â (h1r W2))
//   h2  = g2 @ W2
//   out = log_softmax(h2 + b2)
// GEMMs: V_WMMA_F32_16X16X4_F32 (exact fp32 matrix cores).
// Aggregations: per-(edge,feature) lanes, 16-wide both layers -> 2.5x fewer
// atomics for layer 2 than the naive 40-wide scatter. Tables L2-resident.
// ---------------------------------------------------------------------------

#define F_IN 512
#define HID 16
#define NCLS 40

typedef float v2f __attribute__((ext_vector_type(2)));
typedef float v8f __attribute__((ext_vector_type(8)));

// ---------------- degree / norm ----------------
__global__ void k_deg_init(float* __restrict__ deg, int n) {
  int i = blockIdx.x * blockDim.x + threadIdx.x;
  if (i < n) deg[i] = 1.0f;  // self-loop contributes 1
}

__global__ void k_deg_edges(const int* __restrict__ dst, float* __restrict__ deg, int e) {
  int i = blockIdx.x * blockDim.x + threadIdx.x;
  if (i < e) atomicAdd(&deg[dst[i]], 1.0f);
}

__global__ void k_deg_inv(float* __restrict__ deg, int n) {
  int i = blockIdx.x * blockDim.x + threadIdx.x;
  if (i < n) deg[i] = rsqrtf(deg[i]);  // deg >= 1 always
}

// ---------------- GEMM1: h1[N,16] = x[N,512] @ W1[512,16] (WMMA f32) -------
// One wave per 16-row tile; K in steps of 4.
// A 16x4: lane%16 = row; lanes 0-15 hold K=(k,k+1), lanes 16-31 K=(k+2,k+3).
// B 4x16: mirrored. C 16x16: VGPR j = row j (+8 upper half-wave), lane%16 = col.
__global__ void k_gemm1(const float* __restrict__ x, const float* __restrict__ W1,
                        float* __restrict__ h1, int n) {
  int wave = (blockIdx.x * blockDim.x + threadIdx.x) >> 5;
  int lane = threadIdx.x & 31;
  int m0 = wave * 16;
  if (m0 >= n) return;  // whole-wave exit only (n % 16 == 0)

  int ml = lane & 15;
  int khalf = (lane >> 4) << 1;  // 0 or 2
  const float* __restrict__ xr = x + (size_t)(m0 + ml) * F_IN + khalf;

  v8f acc = {};
#pragma unroll 8
  for (int k = 0; k < F_IN; k += 4) {
    v2f a, b;
    a.x = xr[k];
    a.y = xr[k + 1];
    b.x = W1[(k + khalf) * HID + ml];
    b.y = W1[(k + khalf + 1) * HID + ml];
    acc = __builtin_amdgcn_wmma_f32_16x16x4_f32(false, a, false, b,
                                                (short)0, acc, false, false);
  }

  int mbase = m0 + ((lane >> 4) << 3);
#pragma unroll
  for (int j = 0; j < 8; ++j)
    h1[(size_t)(mbase + j) * HID + ml] = acc[j];
}

// ---------------- 16-wide symmetric-norm aggregation (both layers) ---------
__global__ void k_agg_self16(const float* __restrict__ h, const float* __restrict__ dinv,
                             float* __restrict__ g, int n) {
  int t = blockIdx.x * blockDim.x + threadIdx.x;
  if (t >= n * HID) return;
  int i = t >> 4;
  float d = dinv[i];
  g[t] = h[t] * d * d;  // self-loop term, also zero-initializes accumulator
}

__global__ void k_agg_edges16(const int* __restrict__ src, const int* __restrict__ dst,
                              const float* __restrict__ h, const float* __restrict__ dinv,
                              float* __restrict__ g, int e) {
  int t = blockIdx.x * blockDim.x + threadIdx.x;
  if (t >= e * HID) return;
  int ed = t >> 4;
  int f = t & (HID - 1);
  int s = src[ed], d = dst[ed];
  float nrm = dinv[s] * dinv[d];
  atomicAdd(&g[(size_t)d * HID + f], h[(size_t)s * HID + f] * nrm);
}

// ---------------- bias + relu ----------------------------------------------
__global__ void k_relu_bias(const float* __restrict__ g1, const float* __restrict__ b1,
                            float* __restrict__ h1r, int n) {
  int t = blockIdx.x * blockDim.x + threadIdx.x;
  if (t >= n * HID) return;
  h1r[t] = fmaxf(g1[t] + b1[t & (HID - 1)], 0.0f);
}

// ---------------- GEMM2: h2[N,40] = g2[N,16] @ W2[16,40] (WMMA f32) --------
// 3 column tiles (0,16,32); cols >= 40 zero-padded, stores guarded.
__global__ void k_gemm2(const float* __restrict__ g2, const float* __restrict__ W2,
                        float* __restrict__ h2, int n) {
  int wave = (blockIdx.x * blockDim.x + threadIdx.x) >> 5;
  int lane = threadIdx.x & 31;
  int mt = wave / 3;
  int nt = wave - mt * 3;
  int m0 = mt * 16;
  if (m0 >= n) return;

  int ml = lane & 15;
  int khalf = (lane >> 4) << 1;
  const float* __restrict__ ar = g2 + (size_t)(m0 + ml) * HID + khalf;
  int col = nt * 16 + ml;
  bool colOK = (col < NCLS);

  v8f acc = {};
#pragma unroll
  for (int k = 0; k < HID; k += 4) {
    int ka = k + khalf;
    v2f a, b;
    a.x = ar[k];
    a.y = ar[k + 1];
    b.x = colOK ? W2[ka * NCLS + col] : 0.0f;
    b.y = colOK ? W2[(ka + 1) * NCLS + col] : 0.0f;
    acc = __builtin_amdgcn_wmma_f32_16x16x4_f32(false, a, false, b,
                                                (short)0, acc, false, false);
  }

  if (colOK) {
    int mbase = m0 + ((lane >> 4) << 3);
#pragma unroll
    for (int j = 0; j < 8; ++j)
      h2[(size_t)(mbase + j) * NCLS + col] = acc[j];
  }
}

// ---------------- bias + log_softmax ---------------------------------------
__global__ void k_lsm(const float* __restrict__ h2, const float* __restrict__ b2,
                      float* __restrict__ out, int n) {
  int i = blockIdx.x * blockDim.x + threadIdx.x;
  if (i >= n) return;
  const float* r = h2 + (size_t)i * NCLS;
  float mx = -3.402823466e38f;
#pragma unroll
  for (int c = 0; c < NCLS; ++c) mx = fmaxf(mx, r[c] + b2[c]);
  float s = 0.0f;
#pragma unroll
  for (int c = 0; c < NCLS; ++c) s += __expf(r[c] + b2[c] - mx);
  float lse = mx + __logf(s);
  float* o = out + (size_t)i * NCLS;
#pragma unroll
  for (int c = 0; c < NCLS; ++c) o[c] = r[c] + b2[c] - lse;
}

// ---------------- launcher -------------------------------------------------
extern "C" void kernel_launch(void* const* d_in, const int* in_sizes, int n_in,
                              void* d_out, int out_size, void* d_ws, size_t ws_size,
                              hipStream_t stream) {
  const float* x  = (const float*)d_in[0];
  const int*   ei = (const int*)d_in[1];
  const float* W1 = (const float*)d_in[2];
  const float* b1 = (const float*)d_in[3];
  const float* W2 = (const float*)d_in[4];
  const float* b2 = (const float*)d_in[5];

  int n = in_sizes[0] / F_IN;   // 100000
  int e = in_sizes[1] / 2;      // 3200000
  const int* src = ei;
  const int* dst = ei + e;

  // workspace (floats): dinv(N) | h1(16N) | g1(16N) | h1r(16N) | g2(16N) | h2(40N)
  float* dinv = (float*)d_ws;
  float* h1  = dinv + n;
  float* g1  = h1  + (size_t)HID * n;
  float* h1r = g1  + (size_t)HID * n;
  float* g2  = h1r + (size_t)HID * n;
  float* h2  = g2  + (size_t)HID * n;

  const int B = 256;
  int mtiles = (n + 15) / 16;

  // degrees -> dinv
  k_deg_init<<<(n + B - 1) / B, B, 0, stream>>>(dinv, n);
  k_deg_edges<<<(e + B - 1) / B, B, 0, stream>>>(dst, dinv, e);
  k_deg_inv<<<(n + B - 1) / B, B, 0, stream>>>(dinv, n);

  // layer 1: project, aggregate, activate
  k_gemm1<<<(mtiles * 32 + B - 1) / B, B, 0, stream>>>(x, W1, h1, n);
  k_agg_self16<<<(n * HID + B - 1) / B, B, 0, stream>>>(h1, dinv, g1, n);
  k_agg_edges16<<<(e * HID + B - 1) / B, B, 0, stream>>>(src, dst, h1, dinv, g1, e);
  k_relu_bias<<<(n * HID + B - 1) / B, B, 0, stream>>>(g1, b1, h1r, n);

  // layer 2: aggregate FIRST (16-wide), then project to 40 classes
  k_agg_self16<<<(n * HID + B - 1) / B, B, 0, stream>>>(h1r, dinv, g2, n);
  k_agg_edges16<<<(e * HID + B - 1) / B, B, 0, stream>>>(src, dst, h1r, dinv, g2, e);
  k_gemm2<<<(mtiles * 3 * 32 + B - 1) / B, B, 0, stream>>>(g2, W2, h2, n);

  // bias + log_softmax
  k_lsm<<<(n + B - 1) / B, B, 0, stream>>>(h2, b2, (float*)d_out, n);
}